// Encoder_53369263620316
// MI455X (gfx1250) — compile-verified
//
#include <hip/hip_runtime.h>

// GRU encoder, fused single kernel.
// B=256, T=4096, EMB=32, UNITS=16, gates z/r/h (3*16 = 48 columns).
// One wave32 per 16-row batch tile; fp32 WMMA 16x16x4 for both projections.

typedef float v2f __attribute__((ext_vector_type(2)));
typedef float v8f __attribute__((ext_vector_type(8)));

#define T_STEPS 4096
#define EMB     32
#define NCOL    48   // 3 gates * 16 units

__device__ __forceinline__ float sigmoid_f(float x) {
    return 1.0f / (1.0f + __expf(-x));
}

__global__ __launch_bounds__(32) void gru_fused_kernel(
    const int*   __restrict__ ids,    // [256][4096]
    const float* __restrict__ emb,    // [50001][32]
    const float* __restrict__ wk,     // [32][48]  input kernel
    const float* __restrict__ wr,     // [16][48]  recurrent kernel
    const float* __restrict__ bias,   // [2][48]
    float*       __restrict__ out)    // [256][16]
{
    // LDS transpose buffer for h: C/D layout -> A-operand layout.
    // Row stride 18 floats (72B) keeps (row*18 + even_k)*4 8-byte aligned.
    __shared__ float hbuf[16 * 18];

    const int lane = threadIdx.x & 31;
    const int n    = lane & 15;   // N index (B/C/D layout), also M index (A layout)
    const int half = lane >> 4;   // lane group: selects K-half (A/B), M+8 (C/D)
    const int tile = blockIdx.x;  // 16 tiles of 16 batch rows
    const int brow = tile * 16 + n;

    // ---- Load input-kernel into B-operand registers: bk[gate][kchunk] ----
    // B layout (4x16 f32): lane(n,half), vgpr v -> B[4*kc + 2*half + v][n]
    v2f bk[3][8];
#pragma unroll
    for (int g = 0; g < 3; ++g)
#pragma unroll
        for (int kc = 0; kc < 8; ++kc) {
            const int e = 4 * kc + 2 * half;
            v2f v;
            v.x = wk[(e    ) * NCOL + g * 16 + n];
            v.y = wk[(e + 1) * NCOL + g * 16 + n];
            bk[g][kc] = v;
        }

    // ---- Recurrent kernel into B-operand registers: brr[gate][kchunk] ----
    v2f brr[3][4];
#pragma unroll
    for (int g = 0; g < 3; ++g)
#pragma unroll
        for (int c = 0; c < 4; ++c) {
            const int e = 4 * c + 2 * half;
            v2f v;
            v.x = wr[(e    ) * NCOL + g * 16 + n];
            v.y = wr[(e + 1) * NCOL + g * 16 + n];
            brr[g][c] = v;
        }

    // ---- Biases (broadcast along M in C/D layout: value depends only on n) ----
    float b0[3], b1[3];
#pragma unroll
    for (int g = 0; g < 3; ++g) {
        b0[g] = bias[g * 16 + n];        // input bias
        b1[g] = bias[48 + g * 16 + n];   // recurrent bias
    }

    // ---- Hidden state in C/D layout: h[r] = H[r + 8*half][n], r = 0..7 ----
    v8f h = {};

    // ---- Double-buffered embedding gather in A-operand layout ----
    // A layout (16x4 f32): lane(m=n,half), vgpr v -> A[m][4*kc + 2*half + v]
    const int* idp = ids + (long)brow * T_STEPS;
    v2f a_cur[8], a_nxt[8];
    {
        const int id0 = idp[0];
        const float* er = emb + (long)id0 * EMB + 2 * half;
#pragma unroll
        for (int kc = 0; kc < 8; ++kc)
            a_cur[kc] = *(const v2f*)(er + 4 * kc);   // 8B-aligned b64 loads
    }

    for (int t = 0; t < T_STEPS; ++t) {
        // Prefetch next timestep's embedding slices (hides L2 latency).
        if (t + 1 < T_STEPS) {
            const int id1 = idp[t + 1];
            const float* er = emb + (long)id1 * EMB + 2 * half;
#pragma unroll
            for (int kc = 0; kc < 8; ++kc)
                a_nxt[kc] = *(const v2f*)(er + 4 * kc);
        }

        // --- Transpose h through LDS: C/D layout -> A-operand layout ---
        __syncthreads();   // WAR: previous iter's reads done before overwrite
#pragma unroll
        for (int r = 0; r < 8; ++r)
            hbuf[(r + 8 * half) * 18 + n] = h[r];
        __syncthreads();

        v2f ah[4];
#pragma unroll
        for (int c = 0; c < 4; ++c) {
            const int k = 4 * c + 2 * half;
            v2f v;
            v.x = hbuf[n * 18 + k];
            v.y = hbuf[n * 18 + k + 1];
            ah[c] = v;
        }

        // --- rec = h @ rec_kernel + bias1 : 3 gates x 4 chained WMMAs ---
        v8f rz, rr, rh;
#pragma unroll
        for (int i = 0; i < 8; ++i) { rz[i] = b1[0]; rr[i] = b1[1]; rh[i] = b1[2]; }
#pragma unroll
        for (int c = 0; c < 4; ++c) {
            rz = __builtin_amdgcn_wmma_f32_16x16x4_f32(false, ah[c], false, brr[0][c],
                                                       (short)0, rz, false, false);
            rr = __builtin_amdgcn_wmma_f32_16x16x4_f32(false, ah[c], false, brr[1][c],
                                                       (short)0, rr, false, false);
            rh = __builtin_amdgcn_wmma_f32_16x16x4_f32(false, ah[c], false, brr[2][c],
                                                       (short)0, rh, false, false);
        }

        // --- xp = emb @ kernel + bias0 : 3 gates x 8 chained WMMAs ---
        // (independent of h -> co-executes with the recurrent chain)
        v8f xz, xr, xh;
#pragma unroll
        for (int i = 0; i < 8; ++i) { xz[i] = b0[0]; xr[i] = b0[1]; xh[i] = b0[2]; }
#pragma unroll
        for (int kc = 0; kc < 8; ++kc) {
            xz = __builtin_amdgcn_wmma_f32_16x16x4_f32(false, a_cur[kc], false, bk[0][kc],
                                                       (short)0, xz, false, false);
            xr = __builtin_amdgcn_wmma_f32_16x16x4_f32(false, a_cur[kc], false, bk[1][kc],
                                                       (short)0, xr, false, false);
            xh = __builtin_amdgcn_wmma_f32_16x16x4_f32(false, a_cur[kc], false, bk[2][kc],
                                                       (short)0, xh, false, false);
        }

        // --- gates + state update (elementwise in C/D layout) ---
#pragma unroll
        for (int i = 0; i < 8; ++i) {
            const float z  = sigmoid_f(xz[i] + rz[i]);
            const float r  = sigmoid_f(xr[i] + rr[i]);
            const float hh = sigmoid_f(xh[i] + r * rh[i]);
            h[i] = z * h[i] + (1.0f - z) * hh;
        }

#pragma unroll
        for (int kc = 0; kc < 8; ++kc) a_cur[kc] = a_nxt[kc];
    }

    // ---- Write h_final [256][16] from C/D layout ----
#pragma unroll
    for (int r = 0; r < 8; ++r)
        out[(long)(tile * 16 + r + 8 * half) * 16 + n] = h[r];
}

extern "C" void kernel_launch(void* const* d_in, const int* in_sizes, int n_in,
                              void* d_out, int out_size, void* d_ws, size_t ws_size,
                              hipStream_t stream) {
    (void)in_sizes; (void)n_in; (void)out_size; (void)d_ws; (void)ws_size;
    const int*   ids  = (const int*)d_in[0];
    const float* emb  = (const float*)d_in[1];
    const float* wk   = (const float*)d_in[2];
    const float* wr   = (const float*)d_in[3];
    const float* bias = (const float*)d_in[4];

    // 16 batch tiles of 16 rows, one wave32 each.
    gru_fused_kernel<<<dim3(16), dim3(32), 0, stream>>>(
        ids, emb, wk, wr, bias, (float*)d_out);
}